// MaxSel_64647847740030
// MI455X (gfx1250) — compile-verified
//
#include <hip/hip_runtime.h>
#include <hip/hip_bf16.h>

// MaxSel: max over 4 directional 3x3 Laplacians on [16,3,512,512] f32.
// Memory-bound (AI ~1.9 FLOP/B) -> CDNA5 async global->LDS staging + VALU
// stencil + streaming b128 NT stores. No WMMA on purpose: at 23.3 TB/s the
// kernel is ~5us of pure data movement; matrix pipes are irrelevant here.

namespace {
constexpr int kW       = 512;
constexpr int kH       = 512;
constexpr int kTileH   = 16;                 // output rows per block
constexpr int kLdsRows = kTileH + 2;         // + top/bottom halo
constexpr int kThreads = 256;                // 8 waves (wave32)
constexpr int kTilesPerImg = kH / kTileH;    // 32
}

typedef float f4  __attribute__((ext_vector_type(4)));   // native vec4 float
typedef int   v4i __attribute__((vector_size(16)));      // builtin's elt type

// ---- CDNA5 async global->LDS copy (ASYNCcnt path) ----------------------
__device__ __forceinline__ void async_copy_b128(const float* gsrc, float* ldst) {
#if __has_builtin(__builtin_amdgcn_global_load_async_to_lds_b128)
  __builtin_amdgcn_global_load_async_to_lds_b128(
      (__attribute__((address_space(1))) v4i*)gsrc,
      (__attribute__((address_space(3))) v4i*)ldst,
      0, 0);
#else
  unsigned loff = (unsigned)(unsigned long long)
      (__attribute__((address_space(3))) void*)ldst;     // LDS byte address
  asm volatile("global_load_async_to_lds_b128 %0, %1, off"
               :: "v"(loff),
                  "v"((__attribute__((address_space(1))) const void*)gsrc)
               : "memory");
#endif
}

__device__ __forceinline__ void wait_async_zero() {
#if __has_builtin(__builtin_amdgcn_s_wait_asynccnt)
  __builtin_amdgcn_s_wait_asynccnt(0);
#else
  asm volatile("s_wait_asynccnt 0" ::: "memory");
#endif
}

// ------------------------------------------------------------------------
__global__ __launch_bounds__(kThreads)
void MaxSel_64647847740030_kernel(const float* __restrict__ x,
                                  float* __restrict__ out) {
  __shared__ float tile[kLdsRows][kW];       // 18*512*4 = 36 KB

  const int bid = blockIdx.x;
  const int img = bid / kTilesPerImg;        // 0..47 (B*C images)
  const int r0  = (bid % kTilesPerImg) * kTileH;
  const int tid = threadIdx.x;

  const float* src = x + (size_t)img * (kH * kW);

  // -------- Stage kLdsRows x kW floats: image rows r0-1 .. r0+kTileH ----
  // 18 rows * 128 float4 chunks = 2304 chunks -> 9 per thread.
  constexpr int kChunks = kLdsRows * (kW / 4);
  #pragma unroll
  for (int k = 0; k < kChunks / kThreads; ++k) {
    const int chunk = tid + k * kThreads;
    const int lr = chunk >> 7;               // LDS row 0..17
    const int c0 = (chunk & 127) << 2;       // column (float index)
    const int gr = r0 - 1 + lr;              // source image row
    float* ldst = &tile[lr][c0];
    if ((unsigned)gr < (unsigned)kH) {
      async_copy_b128(src + (size_t)gr * kW + c0, ldst);
    } else {
      *(f4*)ldst = (f4)(0.0f);               // zero pad vertical halo
    }
  }
  wait_async_zero();       // this wave's async copies have landed in LDS
  __syncthreads();         // all waves' copies visible

  // -------- Compute 4 Laplacians + max, 1x4 pixels per strip ------------
  float* dst = out + (size_t)img * (kH * kW) + (size_t)r0 * kW;
  constexpr int kStrips = kTileH * (kW / 4); // 2048 -> 8 per thread
  #pragma unroll
  for (int k = 0; k < kStrips / kThreads; ++k) {
    const int s  = tid + k * kThreads;
    const int r  = s >> 7;                   // output row in tile
    const int c0 = (s & 127) << 2;           // first column of strip

    const float* top = &tile[r][0];          // image row r0+r-1
    const float* mid = &tile[r + 1][0];      // image row r0+r
    const float* bot = &tile[r + 2][0];      // image row r0+r+1

    const f4 t = *(const f4*)(top + c0);
    const f4 m = *(const f4*)(mid + c0);
    const f4 b = *(const f4*)(bot + c0);

    const bool le = (c0 == 0);               // left image edge (zero pad)
    const bool re = (c0 + 4 == kW);          // right image edge
    const float tA[6] = { le ? 0.f : top[c0 - 1], t[0], t[1], t[2], t[3],
                          re ? 0.f : top[c0 + 4] };
    const float mA[6] = { le ? 0.f : mid[c0 - 1], m[0], m[1], m[2], m[3],
                          re ? 0.f : mid[c0 + 4] };
    const float bA[6] = { le ? 0.f : bot[c0 - 1], b[0], b[1], b[2], b[3],
                          re ? 0.f : bot[c0 + 4] };

    f4 o;
    #pragma unroll
    for (int j = 0; j < 4; ++j) {
      const float m2 = -2.0f * mA[j + 1];
      const float hq = mA[j]     + mA[j + 2] + m2;  // [1,-2,1] horizontal
      const float vq = tA[j + 1] + bA[j + 1] + m2;  // vertical
      const float aq = tA[j + 2] + bA[j]     + m2;  // anti-diag: TR + BL
      const float dq = tA[j]     + bA[j + 2] + m2;  // main diag: TL + BR
      o[j] = fmaxf(fmaxf(hq, vq), fmaxf(aq, dq));
    }
    __builtin_nontemporal_store(o, (f4*)(dst + (size_t)r * kW + c0));
  }
}

extern "C" void kernel_launch(void* const* d_in, const int* in_sizes, int n_in,
                              void* d_out, int out_size, void* d_ws, size_t ws_size,
                              hipStream_t stream) {
  (void)n_in; (void)out_size; (void)d_ws; (void)ws_size;
  const float* x = (const float*)d_in[0];
  float* out     = (float*)d_out;
  const int nimg = in_sizes[0] / (kH * kW);            // 16*3 = 48
  const dim3 grid(nimg * kTilesPerImg);                // 1536 blocks
  MaxSel_64647847740030_kernel<<<grid, kThreads, 0, stream>>>(x, out);
}